// GRUModel_15899968930168
// MI455X (gfx1250) — compile-verified
//
#include <hip/hip_runtime.h>

// GRU: B=256, T=1024, I=64, H=128, O=1, fused single kernel.
// grid = 16 workgroups (one 16-row batch tile each), block = 256 threads (8 waves).
// W_hh r-tile in LDS (padded pitch, conflict-free b64); W_hh z/n tiles and all
// W_ih fragments register-cached (extended VGPR file via s_set_vgpr_msb);
// h double-buffered in LDS (1 barrier/step); x A-frags prefetched at step top;
// hardware TRANS ops (v_exp/v_rcp/v_tanh) for gates.
// Matrix ops: V_WMMA_F32_16X16X4_F32 (full fp32 precision).

static constexpr int Bsz = 256;
static constexpr int Tsz = 1024;
static constexpr int Isz = 64;
static constexpr int Hsz = 128;

static constexpr int PWHH = 132;         // W_hh row pitch (dwords): 4-bank row advance
static constexpr int PH   = 132;         // h row pitch
static constexpr int HBUF = 16 * PH;     // one h buffer (dwords)

typedef __attribute__((ext_vector_type(2))) float v2f;
typedef __attribute__((ext_vector_type(8))) float v8f;

#define LOG2E_F 1.4426950408889634f

__device__ __forceinline__ float fast_sigmoid(float v) {
    // 1 / (1 + e^-v) via v_exp_f32 (2^x) + v_rcp_f32
    float e = __builtin_amdgcn_exp2f(v * -LOG2E_F);
    return __builtin_amdgcn_rcpf(1.0f + e);
}

__device__ __forceinline__ float fast_tanh(float v) {
#if __has_builtin(__builtin_amdgcn_tanhf)
    return __builtin_amdgcn_tanhf(v);          // V_TANH_F32
#else
    // tanh(v) = 1 - 2/(2^(2v*log2e) + 1); saturates correctly at +/-inf
    float e = __builtin_amdgcn_exp2f(v * (2.0f * LOG2E_F));
    return 1.0f - 2.0f * __builtin_amdgcn_rcpf(e + 1.0f);
#endif
}

__global__ __launch_bounds__(256) void gru_fused_wmma(
    const float* __restrict__ x,     // [B,T,I]
    const float* __restrict__ W_ih,  // [3H,I]
    const float* __restrict__ W_hh,  // [3H,H]
    const float* __restrict__ b_ih,  // [3H]
    const float* __restrict__ b_hh,  // [3H]
    const float* __restrict__ W_fc,  // [1,H]
    const float* __restrict__ b_fc,  // [1]
    float* __restrict__ out)         // [B,1]
{
    extern __shared__ float lds[];
    float* sWhhR = lds;                        // Hsz * PWHH  (r-gate block only: rows 0..127)
    float* sH    = sWhhR + Hsz * PWHH;         // 2 * HBUF (double-buffered h)

    const int tid  = threadIdx.x;
    const int wave = tid >> 5;                 // 0..7
    const int lane = tid & 31;
    const int hl   = lane >> 4;                // half-wave 0/1
    const int ln   = lane & 15;                // 0..15
    const int kb   = hl * 2;                   // k offset inside a K=4 slab
    const int m0   = blockIdx.x * 16;          // batch-tile base row

    // ---- stage W_hh r-gate rows into LDS with padded pitch; zero both h buffers ----
    for (int idx = tid; idx < Hsz * Hsz; idx += 256) {
        int r = idx >> 7, c = idx & 127;
        sWhhR[r * PWHH + c] = W_hh[idx];
    }
    for (int idx = tid; idx < 2 * HBUF; idx += 256) sH[idx] = 0.0f;

    // wave w owns gate N-tiles: r -> w, z -> w+8, n -> w+16
    const int nR = wave;
    const int nZ = wave + 8;
    const int nN = wave + 16;
    const int col = (nR << 4) + ln;            // 0..127: column inside each gate block
    const float bR  = b_ih[col]           + b_hh[col];
    const float bZ  = b_ih[Hsz + col]     + b_hh[Hsz + col];
    const float bNi = b_ih[2 * Hsz + col];
    const float bNh = b_hh[2 * Hsz + col];

    // ---- register-cache W_ih B-fragments (time-invariant): 96 VGPRs/lane ----
    v2f wiR[16], wiZ[16], wiN[16];
    {
        const float* piR = W_ih + (nR * 16 + ln) * Isz + kb;
        const float* piZ = W_ih + (nZ * 16 + ln) * Isz + kb;
        const float* piN = W_ih + (nN * 16 + ln) * Isz + kb;
        #pragma unroll
        for (int kk = 0; kk < 16; ++kk) {
            wiR[kk] = *(const v2f*)(piR + kk * 4);
            wiZ[kk] = *(const v2f*)(piZ + kk * 4);
            wiN[kk] = *(const v2f*)(piN + kk * 4);
        }
    }

    // ---- register-cache W_hh z- and n-gate tiles (time-invariant): 128 VGPRs/lane ----
    v2f whZ[32], whN[32];
    {
        const float* pZ = W_hh + (nZ * 16 + ln) * Hsz + kb;
        const float* pN = W_hh + (nN * 16 + ln) * Hsz + kb;
        #pragma unroll
        for (int kk = 0; kk < 32; ++kk) {
            whZ[kk] = *(const v2f*)(pZ + kk * 4);
            whN[kk] = *(const v2f*)(pN + kk * 4);
        }
    }

    v8f hreg = {};                             // h tile, D-layout (row j+8*hl, col)

    __syncthreads();

    const float* xbase = x + (size_t)(m0 + ln) * (Tsz * Isz) + kb;

    for (int t = 0; t < Tsz; ++t) {
        // ---- prefetch x A-frags for this step (global latency hidden by W_hh phase) ----
        const float* xt = xbase + t * Isz;
        v2f xa[16];
        #pragma unroll
        for (int kk = 0; kk < 16; ++kk) xa[kk] = *(const v2f*)(xt + kk * 4);

        const float* rb = sH + (t & 1) * HBUF;          // read h_t
        float*       wb = sH + ((t + 1) & 1) * HBUF;    // write h_{t+1}

        v8f accR  = {};
        v8f accZ  = {};
        v8f accNi = {};
        v8f accNh = {};

        // ---- h contribution: [16x128] x [128x384], K = 128 ----
        // r B-frags from LDS; z/n tiles from registers (full unroll required).
        #pragma unroll
        for (int kk = 0; kk < Hsz / 4; ++kk) {
            const int k = kk * 4 + kb;
            v2f a   = *(const v2f*)(&rb[ln * PH + k]);
            v2f bfR = *(const v2f*)(&sWhhR[(nR * 16 + ln) * PWHH + k]);
            accR  = __builtin_amdgcn_wmma_f32_16x16x4_f32(false, a, false, bfR,     (short)0, accR,  false, false);
            accZ  = __builtin_amdgcn_wmma_f32_16x16x4_f32(false, a, false, whZ[kk], (short)0, accZ,  false, false);
            accNh = __builtin_amdgcn_wmma_f32_16x16x4_f32(false, a, false, whN[kk], (short)0, accNh, false, false);
        }

        // ---- x_t contribution: [16x64] x [64x384], K = 64 (pure register WMMA) ----
        #pragma unroll
        for (int kk = 0; kk < 16; ++kk) {
            accR  = __builtin_amdgcn_wmma_f32_16x16x4_f32(false, xa[kk], false, wiR[kk], (short)0, accR,  false, false);
            accZ  = __builtin_amdgcn_wmma_f32_16x16x4_f32(false, xa[kk], false, wiZ[kk], (short)0, accZ,  false, false);
            accNi = __builtin_amdgcn_wmma_f32_16x16x4_f32(false, xa[kk], false, wiN[kk], (short)0, accNi, false, false);
        }

        // ---- gates + state update, D-layout: VGPR j -> row (j + 8*hl), col ----
        // No barrier needed first: writes go to the other h buffer.
        #pragma unroll
        for (int j = 0; j < 8; ++j) {
            float r = fast_sigmoid(accR[j] + bR);
            float z = fast_sigmoid(accZ[j] + bZ);
            float n = fast_tanh(accNi[j] + bNi + r * (accNh[j] + bNh));
            float h = (1.0f - z) * n + z * hreg[j];
            hreg[j] = h;
            wb[(j + 8 * hl) * PH + col] = h;
        }

        __syncthreads();   // wb (= next step's rb) fully written
    }

    // ---- output FC: out[b] = h[b,:] . W_fc + b_fc ----
    // Tsz is even, so the final state lives in buffer 0.
    if (tid < 16) {
        float s = 0.0f;
        for (int c = 0; c < Hsz; ++c) s += sH[tid * PH + c] * W_fc[c];
        out[m0 + tid] = s + b_fc[0];
    }
}

extern "C" void kernel_launch(void* const* d_in, const int* in_sizes, int n_in,
                              void* d_out, int out_size, void* d_ws, size_t ws_size,
                              hipStream_t stream) {
    const float* x    = (const float*)d_in[0];
    const float* W_ih = (const float*)d_in[1];
    const float* W_hh = (const float*)d_in[2];
    const float* b_ih = (const float*)d_in[3];
    const float* b_hh = (const float*)d_in[4];
    const float* W_fc = (const float*)d_in[5];
    const float* b_fc = (const float*)d_in[6];
    float* out = (float*)d_out;

    const size_t shmem = (size_t)(Hsz * PWHH + 2 * HBUF) * sizeof(float); // 84,480 B

    (void)hipFuncSetAttribute((const void*)gru_fused_wmma,
                              hipFuncAttributeMaxDynamicSharedMemorySize, (int)shmem);

    gru_fused_wmma<<<dim3(Bsz / 16), dim3(256), shmem, stream>>>(
        x, W_ih, W_hh, b_ih, b_hh, W_fc, b_fc, out);
}